// forward_warp_80298708566543
// MI455X (gfx1250) — compile-verified
//
#include <hip/hip_runtime.h>

// B=8, C=16, H=512, W=512  (fixed by the reference setup)
#define BB 8
#define CC 16
#define HH 512
#define WW 512
#define HW (HH * WW)

// Native CDNA5 float atomic-add in saddr form: scalar (SGPR-pair) base +
// 32-bit per-lane byte offset, 4 bilinear corners folded into immediate
// offsets (W*4 = 2048 bytes). Non-returning -> STOREcnt, no CAS loop.
__device__ __forceinline__ void corner_atomics(float* base, unsigned voff,
                                               float v_nw, float v_ne,
                                               float v_sw, float v_se) {
    asm volatile(
        "global_atomic_add_f32 %0, %1, %5\n\t"
        "global_atomic_add_f32 %0, %2, %5 offset:4\n\t"
        "global_atomic_add_f32 %0, %3, %5 offset:2048\n\t"
        "global_atomic_add_f32 %0, %4, %5 offset:2052"
        :
        : "v"(voff), "v"(v_nw), "v"(v_ne), "v"(v_sw), "v"(v_se), "s"(base)
        : "memory");
}

// Initialize d_out = im1 (the reference accumulates into im1). float4 -> b128.
__global__ __launch_bounds__(256) void init_out_kernel(
    const float4* __restrict__ im1, float4* __restrict__ out, int n4) {
    int i = blockIdx.x * blockDim.x + threadIdx.x;
    if (i < n4) out[i] = im1[i];
}

__global__ __launch_bounds__(256) void forward_warp_kernel(
    const float* __restrict__ im0,
    const float* __restrict__ flow,
    float* __restrict__ out) {
    // Each block covers 256 consecutive pixels of one (b, h) row:
    // HW = 262144 is divisible by 256, W = 512 = 2 blocks per row,
    // so b and h are block-uniform -> scalar bases, 32-bit lane offsets.
    const int blk = blockIdx.x;
    const int b      = blk >> 10;              // (blk*256) / (H*W)
    const int hw_blk = (blk << 8) & (HW - 1);  // uniform within block
    const int h      = hw_blk >> 9;            // uniform within block
    const int hw     = hw_blk + threadIdx.x;   // per-lane pixel-in-image
    const int w      = hw & (WW - 1);
    const int pix    = (blk << 8) + threadIdx.x;

    // flow is [B,H,W,2] -> contiguous float2 per pixel (coalesced b64 loads)
    float2 f = ((const float2*)flow)[pix];
    float x = (float)w + f.x;
    float y = (float)h + f.y;
    float x0f = floorf(x);
    float y0f = floorf(y);
    int x0 = (int)x0f;
    int y0 = (int)y0f;

    // Scatter only when ALL four corners are in-bounds (matches reference).
    if (x0 < 0 || x0 > WW - 2 || y0 < 0 || y0 > HH - 2) return;

    float fx = x - x0f;
    float fy = y - y0f;
    float gx = 1.0f - fx;
    float gy = 1.0f - fy;
    float w_nw = gx * gy;
    float w_ne = fx * gy;
    float w_sw = gx * fy;
    float w_se = fx * fy;

    // Scalar (block-uniform) bases; per-lane work is a 32-bit offset.
    const float* src_base = im0 + (size_t)b * (CC * HW);
    float*       dst_base = out + (size_t)b * (CC * HW);
    const int    doff     = (y0 << 9) + x0;     // element offset of NW corner

    // Warm the two distinct output cache lines (gfx1250 global_prefetch_b8)
    __builtin_prefetch(dst_base + doff, 0, 0);
    __builtin_prefetch(dst_base + doff + WW, 0, 0);

    // Phase 1: issue all 16 coalesced channel loads (one load clause,
    // single wait), maximizing memory-level parallelism.
    float v[CC];
#pragma unroll
    for (int c = 0; c < CC; ++c) {
        v[c] = src_base[hw + c * HW];
    }

    // Phase 2: stream the 64 corner atomics into L2.
#pragma unroll
    for (int c = 0; c < CC; ++c) {
        corner_atomics(dst_base, (unsigned)((doff + c * HW) << 2),
                       v[c] * w_nw, v[c] * w_ne, v[c] * w_sw, v[c] * w_se);
    }
}

extern "C" void kernel_launch(void* const* d_in, const int* in_sizes, int n_in,
                              void* d_out, int out_size, void* d_ws, size_t ws_size,
                              hipStream_t stream) {
    const float* im0  = (const float*)d_in[0];   // [8,16,512,512]
    const float* flow = (const float*)d_in[1];   // [8,512,512,2]
    const float* im1  = (const float*)d_in[2];   // [8,16,512,512] (zeros)
    float* out = (float*)d_out;                  // [8,16,512,512]

    // 1) out = im1  (also deterministically re-initializes poisoned d_out)
    int n4 = out_size / 4;
    int blocks_init = (n4 + 255) / 256;
    init_out_kernel<<<blocks_init, 256, 0, stream>>>(
        (const float4*)im1, (float4*)out, n4);

    // 2) scatter-add the bilinear splats
    int n_pix = in_sizes[1] / 2;                 // B*H*W = 2,097,152
    int blocks = n_pix / 256;                    // exact multiple
    forward_warp_kernel<<<blocks, 256, 0, stream>>>(im0, flow, out);
}